// SparseCoder_62474594288040
// MI455X (gfx1250) — compile-verified
//
#include <hip/hip_runtime.h>

#define N_TOK 4096
#define KSEL  64
#define DIM   1024
#define NLAT  16384

// GEMM blocking
#define BM   128                 // block tile M
#define BD   128                 // block tile D
#define BK   16                  // K chunk staged in LDS
#define LSTR (BK + 4)            // LDS row stride (floats): 20 -> 80B rows, keeps
                                 // b128 stores / b64 loads aligned, conflict-free

typedef __attribute__((ext_vector_type(2))) float v2f;
typedef __attribute__((ext_vector_type(8))) float v8f;

// ---------------------------------------------------------------------------
// K1: gather-decode.  One block (256 threads) per token row; each thread owns
// 4 consecutive columns -> every k-iteration reads one full 4KB W_dec row
// coalesced (float4 per lane).  Row scalars (idx, act) are block-uniform.
// W_dec (64 MB) sits in the 192 MB L2, so the 1 GB of gathered row reads
// stream at L2 rate.
// ---------------------------------------------------------------------------
__global__ void decode_kernel(const float* __restrict__ latent_acts,
                              const int*   __restrict__ latent_indices,
                              const float* __restrict__ W_dec,
                              const float* __restrict__ b_dec,
                              const float* __restrict__ post_enc,
                              const float* __restrict__ post_enc_scale,
                              float* __restrict__ out) {
    const int n = blockIdx.x;
    const int c = threadIdx.x * 4;            // 256 threads * 4 = 1024 = DIM
    float4 acc = *(const float4*)(b_dec + c);
    #pragma unroll 4
    for (int k = 0; k < KSEL; ++k) {
        const int   idx = latent_indices[n * KSEL + k];
        const float a   = (latent_acts[n * KSEL + k] + post_enc[idx]) * post_enc_scale[idx];
        const float4 w  = *(const float4*)(W_dec + (size_t)idx * DIM + c);
        acc.x += a * w.x;  acc.y += a * w.y;  acc.z += a * w.z;  acc.w += a * w.w;
    }
    *(float4*)(out + (size_t)n * DIM + c) = acc;
}

// ---------------------------------------------------------------------------
// K2: skip GEMM  out[n,d] += sum_j x[n,j] * W_skip[d,j]
//
// LDS-staged, register-blocked V_WMMA_F32_16X16X4_F32:
//   block (8 waves, 256 thr) -> 128x128 output tile
//   wave grid 2(M) x 4(D)    -> each wave 64x32 = 4x2 WMMA accumulators
//   K staged in 16-wide chunks: A_lds[128][20], B_lds[128][20] = 20 KB LDS
//
// WMMA f32 lane layouts (ISA 7.12.2):
//   A 16x4: lane l -> M = l%16, K = {0,1} + 2*(l/16)   (contiguous float2)
//   B 4x16: lane l -> N = l%16, K = {0,1} + 2*(l/16)   (contiguous float2)
//   C/D:    VGPR r -> M = r + 8*(l/16), N = l%16
// Both x and W_skip are row-major with K as the fast axis, so the staging
// copy and the fragment reads are all contiguous float4 / float2 accesses.
// ---------------------------------------------------------------------------
__global__ void skip_gemm_kernel(const float* __restrict__ x,
                                 const float* __restrict__ W_skip,
                                 float* __restrict__ out) {
    __shared__ float A_lds[BM * LSTR];
    __shared__ float B_lds[BD * LSTR];

    const int tid  = threadIdx.x;
    const int lane = tid & 31;
    const int wave = tid >> 5;                 // 0..7
    const int wm   = wave >> 2;                // 0..1 : M strip (64 rows)
    const int wd   = wave & 3;                 // 0..3 : D strip (32 cols)

    const int bm = blockIdx.y * BM;            // 32 m-blocks
    const int bd = blockIdx.x * BD;            // 8  d-blocks

    const int l16   = lane & 15;
    const int lhi   = lane >> 4;               // 0/1
    const int khalf = lhi * 2;                 // 0 or 2

    // staging map: 512 float4 per operand per chunk, 2 per thread
    const int srow0 = tid >> 2;                // rows 0..63   (pass 0)
    const int srow1 = srow0 + 64;              // rows 64..127 (pass 1)
    const int scol  = (tid & 3) * 4;           // 0,4,8,12

    v8f acc[4][2];
    #pragma unroll
    for (int mi = 0; mi < 4; ++mi)
        #pragma unroll
        for (int di = 0; di < 2; ++di)
            acc[mi][di] = (v8f){0.f,0.f,0.f,0.f,0.f,0.f,0.f,0.f};

    for (int k0 = 0; k0 < DIM; k0 += BK) {
        // ---- stage A (x rows) and B (W_skip rows) into LDS ----
        {
            const float4 a0 = *(const float4*)(x      + (size_t)(bm + srow0) * DIM + k0 + scol);
            const float4 a1 = *(const float4*)(x      + (size_t)(bm + srow1) * DIM + k0 + scol);
            const float4 b0 = *(const float4*)(W_skip + (size_t)(bd + srow0) * DIM + k0 + scol);
            const float4 b1 = *(const float4*)(W_skip + (size_t)(bd + srow1) * DIM + k0 + scol);
            *(float4*)(A_lds + srow0 * LSTR + scol) = a0;
            *(float4*)(A_lds + srow1 * LSTR + scol) = a1;
            *(float4*)(B_lds + srow0 * LSTR + scol) = b0;
            *(float4*)(B_lds + srow1 * LSTR + scol) = b1;
        }
        __syncthreads();

        // ---- 4 WMMA K-steps over the chunk, 8 WMMAs each ----
        #pragma unroll
        for (int kk = 0; kk < BK; kk += 4) {
            v2f af[4], bf[2];
            #pragma unroll
            for (int mi = 0; mi < 4; ++mi)
                af[mi] = *(const v2f*)(A_lds + (wm * 64 + mi * 16 + l16) * LSTR + kk + khalf);
            #pragma unroll
            for (int di = 0; di < 2; ++di)
                bf[di] = *(const v2f*)(B_lds + (wd * 32 + di * 16 + l16) * LSTR + kk + khalf);
            #pragma unroll
            for (int mi = 0; mi < 4; ++mi)
                #pragma unroll
                for (int di = 0; di < 2; ++di)
                    acc[mi][di] = __builtin_amdgcn_wmma_f32_16x16x4_f32(
                        false, af[mi], false, bf[di], (short)0, acc[mi][di], false, false);
        }
        __syncthreads();
    }

    // ---- epilogue: RMW onto decode result ----
    #pragma unroll
    for (int mi = 0; mi < 4; ++mi) {
        const int nbase = bm + wm * 64 + mi * 16 + 8 * lhi;
        #pragma unroll
        for (int di = 0; di < 2; ++di) {
            const int dcol = bd + wd * 32 + di * 16 + l16;
            #pragma unroll
            for (int r = 0; r < 8; ++r) {
                const size_t o = (size_t)(nbase + r) * DIM + dcol;
                out[o] += acc[mi][di][r];
            }
        }
    }
}

// ---------------------------------------------------------------------------
// K3a: ws[2] = sum_d (sum_n y[n,d])^2   (4 blocks x 256 thr, coalesced rows)
// ---------------------------------------------------------------------------
__global__ void colsumsq_kernel(const float* __restrict__ y, float* __restrict__ ws) {
    const int d = blockIdx.x * blockDim.x + threadIdx.x;   // 0..1023
    float s = 0.f;
    for (int n = 0; n < N_TOK; ++n) s += y[(size_t)n * DIM + d];
    __shared__ float red[256];
    red[threadIdx.x] = s * s;
    __syncthreads();
    for (int off = 128; off > 0; off >>= 1) {
        if (threadIdx.x < off) red[threadIdx.x] += red[threadIdx.x + off];
        __syncthreads();
    }
    if (threadIdx.x == 0) atomicAdd(&ws[2], red[0]);
}

// ---------------------------------------------------------------------------
// K3b: ws[0] += sum (y - out)^2 ;  ws[1] += sum y^2
// ---------------------------------------------------------------------------
__global__ void errsum_kernel(const float* __restrict__ y,
                              const float* __restrict__ out,
                              float* __restrict__ ws) {
    size_t i            = (size_t)blockIdx.x * blockDim.x + threadIdx.x;
    const size_t stride = (size_t)gridDim.x * blockDim.x;
    float se = 0.f, sy2 = 0.f;
    for (; i < (size_t)N_TOK * DIM; i += stride) {
        const float yv = y[i];
        const float e  = yv - out[i];
        se  += e * e;
        sy2 += yv * yv;
    }
    __shared__ float r0[256];
    __shared__ float r1[256];
    r0[threadIdx.x] = se;
    r1[threadIdx.x] = sy2;
    __syncthreads();
    for (int off = 128; off > 0; off >>= 1) {
        if (threadIdx.x < off) {
            r0[threadIdx.x] += r0[threadIdx.x + off];
            r1[threadIdx.x] += r1[threadIdx.x + off];
        }
        __syncthreads();
    }
    if (threadIdx.x == 0) {
        atomicAdd(&ws[0], r0[0]);
        atomicAdd(&ws[1], r1[0]);
    }
}

// ---------------------------------------------------------------------------
// K4: fvu = err / (sum y^2 - (sum_d colsum_d^2)/N)
// ---------------------------------------------------------------------------
__global__ void finalize_kernel(const float* __restrict__ ws, float* __restrict__ fvu_out) {
    const float tv = ws[1] - ws[2] / (float)N_TOK;
    *fvu_out = ws[0] / tv;
}

// ---------------------------------------------------------------------------
// Inputs (setup_inputs order):
//  0 x[N,D] f32      1 y[N,D] f32        2 latent_acts[N,K] f32
//  3 latent_indices[N,K] i32             4 W_dec[L,D] f32
//  5 b_dec[D] f32    6 post_enc[L] f32   7 post_enc_scale[L] f32
//  8 W_skip[D,D] f32
// d_out: N*D floats (sae_out) followed by 1 float (fvu).
// ---------------------------------------------------------------------------
extern "C" void kernel_launch(void* const* d_in, const int* in_sizes, int n_in,
                              void* d_out, int out_size, void* d_ws, size_t ws_size,
                              hipStream_t stream) {
    const float* x        = (const float*)d_in[0];
    const float* y        = (const float*)d_in[1];
    const float* lat_acts = (const float*)d_in[2];
    const int*   lat_idx  = (const int*)  d_in[3];
    const float* W_dec    = (const float*)d_in[4];
    const float* b_dec    = (const float*)d_in[5];
    const float* post_enc = (const float*)d_in[6];
    const float* post_sc  = (const float*)d_in[7];
    const float* W_skip   = (const float*)d_in[8];

    float* out = (float*)d_out;             // [N*D] sae_out, then [1] fvu
    float* ws  = (float*)d_ws;              // ws[0]=err ws[1]=sumY2 ws[2]=sumColSumSq

    hipMemsetAsync(d_ws, 0, 4 * sizeof(float), stream);

    decode_kernel<<<N_TOK, 256, 0, stream>>>(lat_acts, lat_idx, W_dec, b_dec,
                                             post_enc, post_sc, out);

    // 128x128 block tiles: grid (D/BD, N/BM) = (8, 32)
    dim3 ggrid(DIM / BD, N_TOK / BM);
    skip_gemm_kernel<<<ggrid, 256, 0, stream>>>(x, W_skip, out);

    colsumsq_kernel<<<DIM / 256, 256, 0, stream>>>(y, ws);
    errsum_kernel<<<512, 256, 0, stream>>>(y, out, ws);
    finalize_kernel<<<1, 1, 0, stream>>>(ws, out + (size_t)N_TOK * DIM);
}